// LayerModule_19104014533127
// MI455X (gfx1250) — compile-verified
//
#include <hip/hip_runtime.h>

// ---------------- problem constants ----------------
#define J_  64
#define T_  64
#define NV  4
#define KC  8
#define I_  64
#define M_  (2 * J_ * T_ + 1)          // 8193 terms per node
#define BLK (NV * KC)                  // 32 floats = 128 B per (i,term) block
#define ROWSTRIDE ((size_t)M_ * BLK)   // 262176 floats per i
#define UNDER_FLOATS ((size_t)I_ * ROWSTRIDE)  // 16,779,264

typedef float v8f __attribute__((ext_vector_type(8)));
typedef float v2f __attribute__((ext_vector_type(2)));
typedef float f4  __attribute__((ext_vector_type(4)));
typedef int   v4i __attribute__((ext_vector_type(4)));

#define ASG __attribute__((address_space(1)))
#define ASL __attribute__((address_space(3)))

#if defined(__has_builtin)
# if __has_builtin(__builtin_amdgcn_wmma_f32_16x16x4_f32)
#  define HAS_WMMA4 1
# endif
# if __has_builtin(__builtin_amdgcn_global_load_async_to_lds_b128)
#  define HAS_ASYNC 1
# endif
# if __has_builtin(__builtin_amdgcn_s_wait_asynccnt)
#  define HAS_WAITASYNC 1
# endif
#endif
#ifndef HAS_WMMA4
#define HAS_WMMA4 0
#endif
#ifndef HAS_ASYNC
#define HAS_ASYNC 0
#endif
#ifndef HAS_WAITASYNC
#define HAS_WAITASYNC 0
#endif

// ---------------------------------------------------------------------------
// Kernel 1: broadcast of variable rows 1..3 (75% of output bytes).
// Value is independent of i and of under/over -> keep one float4 in a register
// and stream 32 NT b128 stores.  grid = (J*4 t-chunks, 4 i-chunks) x 192 thr.
// ---------------------------------------------------------------------------
__global__ __launch_bounds__(192) void
bcast_var123(const float* __restrict__ liu, const float* __restrict__ lio,
             float* __restrict__ outU, float* __restrict__ outO) {
    const int j   = blockIdx.x >> 2;
    const int t0  = (blockIdx.x & 3) << 4;      // 16 t's per block
    const int i0  = blockIdx.y << 4;            // 16 i's per block
    const int s   = threadIdx.x;                // 0..191
    const int part = s / 96;                    // 0: lio terms, 1: liu terms
    const int r    = s % 96;
    const int tl   = r / 6;                     // local t (0..15)
    const int q    = r % 6;                     // float4 slot within var1..3 (96B)

    const float* src = part ? liu : lio;
    const f4 val = *(const f4*)(src + (size_t)((j * T_ + t0 + tl) * BLK + 8 + q * 4));

    const int term = part * (J_ * T_) + j * T_ + t0 + tl;
    const size_t base = (size_t)term * BLK + 8 + q * 4;

    #pragma unroll 4
    for (int i = i0; i < i0 + 16; ++i) {
        const size_t off = (size_t)i * ROWSTRIDE + base;
        __builtin_nontemporal_store(val, (f4*)(outU + off));
        __builtin_nontemporal_store(val, (f4*)(outO + off));
    }
}

// ---------------------------------------------------------------------------
// Kernel 2: var==0 rows via V_WMMA_F32_16X16X4_F32 outer products.
// D(M=i_local, N=col) = A(M,K=0)*B(K=0,N) with A = w_sel[i,j], B = v[t,k].
// var0 source rows (1 KB/block) staged to LDS with async-to-LDS.
// grid = 256 blocks (j, t-chunk) x 256 threads (8 waves).
// wave w: combo = w%4 in {under-lio(w_neg), under-liu(w_pos),
//                         over-lio(w_pos),  over-liu(w_neg)},
//         i-tiles {0,1} or {2,3} from w/4.
// ---------------------------------------------------------------------------
__global__ __launch_bounds__(256) void
wmma_var0(const float* __restrict__ liu, const float* __restrict__ lio,
          const float* __restrict__ W,
          float* __restrict__ outU, float* __restrict__ outO) {
    __shared__ float vlds[256];                 // [part][tl*8 + k], 1 KB
    const int j   = blockIdx.x >> 2;
    const int t0  = (blockIdx.x & 3) << 4;
    const int tid = threadIdx.x;

    if (tid < 64) {                             // 64 lanes x 16 B = 1 KB stage
        const int part = tid >> 5;
        const int tt   = (tid & 31) >> 1;
        const int half = tid & 1;
        const float* g = (part ? liu : lio) +
                         (size_t)((j * T_ + t0 + tt) * BLK + half * 4);
        const int ldsOff = part * 128 + tt * 8 + half * 4;
#if HAS_ASYNC
        __builtin_amdgcn_global_load_async_to_lds_b128(
            (ASG v4i*)g, (ASL v4i*)&vlds[ldsOff], 0, 0);
#else
        *(f4*)&vlds[ldsOff] = *(const f4*)g;
#endif
    }
#if HAS_WAITASYNC
    __builtin_amdgcn_s_wait_asynccnt(0);
#else
    asm volatile("s_wait_asynccnt 0" ::: "memory");
#endif
    __syncthreads();

    const int w      = tid >> 5;
    const int lane   = tid & 31;
    const int combo  = w & 3;
    const int itbase = (w >> 2) << 1;           // 0 or 2
    const int part   = combo & 1;               // 0: lio source, 1: liu source
    const bool neg   = (combo == 0) || (combo == 3);
    float* outBase   = (combo < 2) ? outU : outO;

    // A operands: lanes 0-15 carry K=0 column = w_sel[i0+lane, j]; rest zero.
    v2f a[2];
    #pragma unroll
    for (int ii = 0; ii < 2; ++ii) {
        const int i0 = (itbase + ii) << 4;
        const float x  = W[(size_t)(i0 + (lane & 15)) * J_ + j];
        const float ws = neg ? fminf(x, 0.0f) : fmaxf(x, 0.0f);
        a[ii] = (v2f){(lane < 16) ? ws : 0.0f, 0.0f};
    }

    const int hi = lane >> 4;                   // D half: rows r or r+8
    const int cl = lane & 15;                   // D column
    #pragma unroll
    for (int ct = 0; ct < 8; ++ct) {            // 8 column tiles of 16 (= 16t x 8k)
        const float bvv = vlds[part * 128 + ct * 16 + (lane & 15)];
        const v2f b = (v2f){(lane < 16) ? bvv : 0.0f, 0.0f};

        const int c2   = ct * 16 + cl;          // this lane's D column
        const int tl   = c2 >> 3;
        const int kk   = c2 & 7;
        const int term = part * (J_ * T_) + j * T_ + t0 + tl;

        #pragma unroll
        for (int ii = 0; ii < 2; ++ii) {
            v8f d;
#if HAS_WMMA4
            v8f c0 = {};
            d = __builtin_amdgcn_wmma_f32_16x16x4_f32(
                    false, a[ii], false, b, (short)0, c0, false, false);
#else
            {   // scalar fallback (same values, no matrix unit)
                const float bcol = vlds[part * 128 + ct * 16 + cl];
                #pragma unroll
                for (int r = 0; r < 8; ++r) {
                    const float x = W[(size_t)(((itbase + ii) << 4) + hi * 8 + r) * J_ + j];
                    d[r] = (neg ? fminf(x, 0.0f) : fmaxf(x, 0.0f)) * bcol;
                }
            }
#endif
            const int i0 = ((itbase + ii) << 4) + hi * 8;   // + r below
            const size_t base = (size_t)i0 * ROWSTRIDE + (size_t)term * BLK + kk;
            #pragma unroll
            for (int r = 0; r < 8; ++r)
                __builtin_nontemporal_store(d[r], outBase + base + (size_t)r * ROWSTRIDE);
        }
    }
}

// ---------------------------------------------------------------------------
// Kernel 3: bias term (term == 8192) + degree outputs (max over j, per var).
// ---------------------------------------------------------------------------
__global__ __launch_bounds__(256) void
tail(const float* __restrict__ ud, const float* __restrict__ od,
     const float* __restrict__ bias,
     float* __restrict__ outU, float* __restrict__ outO,
     float* __restrict__ udeg, float* __restrict__ odeg) {
    __shared__ float m[256];
    const int tid = threadIdx.x;

    for (int idx = tid; idx < I_ * BLK; idx += 256) {
        const int i = idx >> 5, p = idx & 31;
        const float val = (p < 8) ? bias[i] : 1.0f;
        const size_t off = (size_t)i * ROWSTRIDE + (size_t)(M_ - 1) * BLK + p;
        outU[off] = val;
        outO[off] = val;
    }

    m[tid] = fmaxf(ud[tid], od[tid]);           // flat [j*4 + var]
    __syncthreads();
    #pragma unroll
    for (int s = 128; s >= 4; s >>= 1) {        // reduce over j, keep var lanes
        if (tid < s) m[tid] = fmaxf(m[tid], m[tid + s]);
        __syncthreads();
    }
    const float dv = m[tid & 3];                // deg[var]
    udeg[tid] = dv;                             // [I=64, NV=4] broadcast
    odeg[tid] = dv;
}

// ---------------------------------------------------------------------------
extern "C" void kernel_launch(void* const* d_in, const int* in_sizes, int n_in,
                              void* d_out, int out_size, void* d_ws, size_t ws_size,
                              hipStream_t stream) {
    const float* liu  = (const float*)d_in[0];   // layer_inputs_under [J,T,4,8]
    const float* lio  = (const float*)d_in[1];   // layer_inputs_over  [J,T,4,8]
    const float* ud   = (const float*)d_in[2];   // under degrees [J,4]
    const float* od   = (const float*)d_in[3];   // over degrees  [J,4]
    const float* W    = (const float*)d_in[4];   // weights [I,J]
    const float* bias = (const float*)d_in[5];   // biases [I]

    float* outU = (float*)d_out;
    float* outO = outU + UNDER_FLOATS;
    float* udeg = outO + UNDER_FLOATS;
    float* odeg = udeg + (size_t)I_ * NV;

    bcast_var123<<<dim3(J_ * 4, 4), 192, 0, stream>>>(liu, lio, outU, outO);
    wmma_var0  <<<dim3(J_ * 4),     256, 0, stream>>>(liu, lio, W, outU, outO);
    tail       <<<dim3(1),          256, 0, stream>>>(ud, od, bias, outU, outO, udeg, odeg);
}